// StochasticRGNLoss_83056077570644
// MI455X (gfx1250) — compile-verified
//
#include <hip/hip_runtime.h>

// ---------------- problem constants (from the reference) ----------------
#define GROUPS   64
#define LPTS     4096
#define NPTS     (GROUPS * LPTS)      // 262144 points
#define NPAIR    (GROUPS * 65536)     // 4194304 pairs
#define BLOCKS   2048
#define TPB      256
#define NTHREADS (BLOCKS * TPB)       // 524288
#define ITERS    (NPAIR / NTHREADS)   // 8, exact -> no divergence

typedef __attribute__((ext_vector_type(2))) float v2f;
typedef __attribute__((ext_vector_type(8))) float v8f;

__device__ __forceinline__ float readlane_f32(float v, int lane) {
    return __int_as_float(__builtin_amdgcn_readlane(__float_as_int(v), lane));
}

// Sum 32 per-lane values across the wave using V_WMMA_F32_16X16X4_F32.
// A (16x4 f32, 2 VGPRs): VGPR0 lanes0-15 -> (M=lane,K=0), lanes16-31 -> (M=lane-16,K=2).
// Lane value goes in A-VGPR0, zero in A-VGPR1, B = all ones (4x16).
// D[m][n] = val[m] + val[m+16]; D VGPR r holds rowsum_r in lanes0-15 and
// rowsum_{r+8} in lanes16-31, so total = sum_r (lane0(d[r]) + lane16(d[r])).
// Extraction via v_readlane_b32 (constant lanes) -> no LDS traffic.
// NOTE: requires EXEC == all ones (callers have no divergence at this point).
__device__ __forceinline__ float wave_reduce_wmma(float v) {
    v2f a; a[0] = v;    a[1] = 0.0f;
    v2f b; b[0] = 1.0f; b[1] = 1.0f;
    v8f c = {};
    v8f d = __builtin_amdgcn_wmma_f32_16x16x4_f32(
        /*neg_a=*/false, a, /*neg_b=*/false, b,
        /*c_mod=*/(short)0, c, /*reuse_a=*/false, /*reuse_b=*/false);
    float s = 0.0f;
#pragma unroll
    for (int r = 0; r < 8; ++r)
        s += readlane_f32(d[r], 0) + readlane_f32(d[r], 16);
    return s;
}

// Kernel 0: pack CA rows (inputs[:,1,:], target[:,1,:]) into an aligned
// float4 table in workspace so the hot-loop gathers are single b128 loads
// out of L2 (whole table = 8MB, fully L2-resident on 192MB L2).
__global__ void rgn_pack_ca(const float* __restrict__ inp,
                            const float* __restrict__ tgt,
                            float4* __restrict__ ca) {
    int i = blockIdx.x * blockDim.x + threadIdx.x;   // grid sized exactly
    const float* p = inp + (size_t)i * 9 + 3;
    const float* q = tgt + (size_t)i * 9 + 3;
    ca[i]        = make_float4(p[0], p[1], p[2], 0.0f);
    ca[NPTS + i] = make_float4(q[0], q[1], q[2], 0.0f);
}

// Kernel 1: grid-stride (exact multiple -> no bounds divergence) over pairs.
// Streams left/right int64 indices coalesced, gathers CA coords, accumulates
// (|din| - |dtg|)^2 per thread, then WMMA wave reduce + LDS block reduce.
__global__ void rgn_main(const float* __restrict__ inp,
                         const float* __restrict__ tgt,
                         const long long* __restrict__ left,
                         const long long* __restrict__ right,
                         const float4* __restrict__ ca,   // may be null -> slow path
                         float* __restrict__ partials) {
    const int t = blockIdx.x * TPB + threadIdx.x;
    float acc = 0.0f;
#pragma unroll
    for (int j = 0; j < ITERS; ++j) {
        const int p = t + j * NTHREADS;
        if (j + 1 < ITERS) {
            // gfx1250: lowers to global_prefetch_b8 (speculative, RT)
            __builtin_prefetch(left  + p + NTHREADS, 0, 0);
            __builtin_prefetch(right + p + NTHREADS, 0, 0);
        }
        const int l = (int)left[p];
        const int r = (int)right[p];
        float dx, dy, dz, ex, ey, ez;
        if (ca) {
            float4 al = ca[l],        ar = ca[r];
            float4 bl = ca[NPTS + l], br = ca[NPTS + r];
            dx = al.x - ar.x; dy = al.y - ar.y; dz = al.z - ar.z;
            ex = bl.x - br.x; ey = bl.y - br.y; ez = bl.z - br.z;
        } else {
            const float* pl = inp + (size_t)l * 9 + 3;
            const float* pr = inp + (size_t)r * 9 + 3;
            dx = pl[0] - pr[0]; dy = pl[1] - pr[1]; dz = pl[2] - pr[2];
            const float* ql = tgt + (size_t)l * 9 + 3;
            const float* qr = tgt + (size_t)r * 9 + 3;
            ex = ql[0] - qr[0]; ey = ql[1] - qr[1]; ez = ql[2] - qr[2];
        }
        // raw v_sqrt_f32 (~1 ulp) instead of the correctly-rounded fixup chain
        const float din = __builtin_amdgcn_sqrtf(dx * dx + dy * dy + dz * dz);
        const float dtg = __builtin_amdgcn_sqrtf(ex * ex + ey * ey + ez * ez);
        const float diff = din - dtg;
        acc += diff * diff;
    }

    // All 32 lanes active here -> EXEC all-1s for WMMA.
    const float wsum = wave_reduce_wmma(acc);

    __shared__ float ws[TPB / 32];
    const int lane = threadIdx.x & 31;
    const int wid  = threadIdx.x >> 5;
    if (lane == 0) ws[wid] = wsum;
    __syncthreads();
    if (threadIdx.x == 0) {
        float s = 0.0f;
#pragma unroll
        for (int i = 0; i < TPB / 32; ++i) s += ws[i];
        partials[blockIdx.x] = s;
    }
}

// Kernel 2: single wave, deterministic fixed-order fold of the 2048 block
// partials (BLOCKS % 32 == 0 -> uniform loop, EXEC stays all-1s for WMMA).
__global__ void rgn_final(const float* __restrict__ partials,
                          float* __restrict__ out) {
    float acc = 0.0f;
    for (int i = threadIdx.x; i < BLOCKS; i += 32) acc += partials[i];
    const float total = wave_reduce_wmma(acc);
    if (threadIdx.x == 0) out[0] = total * (1.0f / (float)NPAIR);
}

extern "C" void kernel_launch(void* const* d_in, const int* in_sizes, int n_in,
                              void* d_out, int out_size, void* d_ws, size_t ws_size,
                              hipStream_t stream) {
    const float*     inp   = (const float*)d_in[0];      // (N,3,3) f32
    const float*     tgt   = (const float*)d_in[1];      // (N,3,3) f32
    /* d_in[2] = mask — unused by the reference math */
    const long long* left  = (const long long*)d_in[3];  // (P,) int64
    const long long* right = (const long long*)d_in[4];  // (P,) int64
    float* out = (float*)d_out;

    // workspace layout: [0, 8KB) block partials | [8KB, ...) packed CA table
    float* partials = (float*)d_ws;
    const size_t packed_off = 8192;                       // 16B aligned
    const size_t need = packed_off + (size_t)2 * NPTS * sizeof(float4); // ~8.4MB
    float4* ca = nullptr;
    if (ws_size >= need) {
        ca = (float4*)((char*)d_ws + packed_off);
        rgn_pack_ca<<<NPTS / 256, 256, 0, stream>>>(inp, tgt, ca);
    }

    rgn_main<<<BLOCKS, TPB, 0, stream>>>(inp, tgt, left, right, ca, partials);
    rgn_final<<<1, 32, 0, stream>>>(partials, out);
}